// GMM_39032662786561
// MI455X (gfx1250) — compile-verified
//
#include <hip/hip_runtime.h>
#include <hip/hip_bf16.h>
#include <utility>

typedef __attribute__((ext_vector_type(2))) float v2f;
typedef __attribute__((ext_vector_type(8))) float v8f;

// ---------------------------------------------------------------------------
// Feature index mapping: j in [0,136) -> upper-triangular pair (d<=e),
// j in [136,152) -> linear term (x_d * 1), j >= 152 -> constant / padding.
// xv[16] holds 1.0f so every feature is xv[d]*xv[e].
// ---------------------------------------------------------------------------
__device__ __host__ constexpr int pair_d(int j) {
    if (j >= 152) return 16;
    if (j >= 136) return j - 136;
    int c = 0;
    for (int d = 0; d < 16; ++d) {
        int row = 16 - d;
        if (j < c + row) return d;
        c += row;
    }
    return 16;
}
__device__ __host__ constexpr int pair_e(int j) {
    if (j >= 152) return 16;
    if (j >= 136) return 16;
    int c = 0;
    for (int d = 0; d < 16; ++d) {
        int row = 16 - d;
        if (j < c + row) return d + (j - c);
        c += row;
    }
    return 16;
}

// One WMMA step: build the 2 A-operand values for this lane (A layout:
// lanes 0-15 supply K = 4S,4S+1 ; lanes 16-31 supply K = 4S+2,4S+3).
template <int S>
__device__ __forceinline__ v8f step(const float* xv, bool hi, const v2f* b, v8f c) {
    float f0 = xv[pair_d(4 * S + 0)] * xv[pair_e(4 * S + 0)];
    float f1 = xv[pair_d(4 * S + 1)] * xv[pair_e(4 * S + 1)];
    float f2 = xv[pair_d(4 * S + 2)] * xv[pair_e(4 * S + 2)];
    float f3 = xv[pair_d(4 * S + 3)] * xv[pair_e(4 * S + 3)];
    v2f a;
    a.x = hi ? f2 : f0;
    a.y = hi ? f3 : f1;
    // D = A(16x4) * B(4x16) + C, fp32 WMMA
    return __builtin_amdgcn_wmma_f32_16x16x4_f32(false, a, false, b[S], (short)0, c,
                                                 false, false);
}

template <int... S>
__device__ __forceinline__ v8f run_steps(std::integer_sequence<int, S...>,
                                         const float* xv, bool hi, const v2f* b, v8f c) {
    ((c = step<S>(xv, hi, b, c)), ...);
    return c;
}

// ---------------------------------------------------------------------------
// Prep: one thread per mixture. Invert Sigma_k (Gauss-Jordan, SPD so no
// pivoting), det from pivots, fold Phi / det / mu terms into W[156][16]:
//   g_{n,k} = sum_j f_j(x_n) * W[j][k]
//           = -0.5 x' S^-1 x + (S^-1 mu)'x + (-0.5 mu'S^-1 mu + log Phi
//             - 0.5 log(2 pi det))
// ---------------------------------------------------------------------------
__global__ void gmm_prep(const float* __restrict__ Phi, const float* __restrict__ mu,
                         const float* __restrict__ Sigma, float* __restrict__ W) {
    int k = threadIdx.x;
    if (k >= 16) return;
    float A[16][32];
    for (int i = 0; i < 16; ++i)
        for (int j = 0; j < 16; ++j) {
            A[i][j] = Sigma[k * 256 + i * 16 + j];
            A[i][16 + j] = (i == j) ? 1.f : 0.f;
        }
    float det = 1.f;
    for (int p = 0; p < 16; ++p) {
        float piv = A[p][p];
        det *= piv;
        float inv = 1.f / piv;
        for (int j = 0; j < 32; ++j) A[p][j] *= inv;
        for (int r = 0; r < 16; ++r) {
            if (r == p) continue;
            float f = A[r][p];
            for (int j = 0; j < 32; ++j) A[r][j] -= f * A[p][j];
        }
    }
    float m[16], y[16];
    for (int d = 0; d < 16; ++d) m[d] = mu[k * 16 + d];
    float cquad = 0.f;
    for (int d = 0; d < 16; ++d) {
        float s = 0.f;
        for (int e = 0; e < 16; ++e) s += A[d][16 + e] * m[e];
        y[d] = s;
        cquad += m[d] * s;
    }
    int j = 0;
    for (int d = 0; d < 16; ++d)
        for (int e = d; e < 16; ++e) {
            float s = (d == e) ? A[d][16 + d] : (A[d][16 + e] + A[e][16 + d]);
            W[j * 16 + k] = -0.5f * s;
            ++j;
        }
    for (int d = 0; d < 16; ++d) W[(136 + d) * 16 + k] = y[d];
    W[152 * 16 + k] = -0.5f * cquad + __logf(Phi[k])
                    - 0.5f * __logf(2.f * 3.14159265358979f * det);
    W[153 * 16 + k] = 0.f;
    W[154 * 16 + k] = 0.f;
    W[155 * 16 + k] = 0.f;
}

// ---------------------------------------------------------------------------
// Main: each wave handles 16-sample tiles. B operand (W) staged LDS -> VGPRs
// once, reused across the grid-stride tile loop. 39 WMMAs per tile produce
// the 16x16 (sample x mixture) log-density tile in C layout; epilogue does
// exp + 16-lane xor reduction over mixtures + -log + store.
// ---------------------------------------------------------------------------
__global__ __launch_bounds__(256) void gmm_main(const float* __restrict__ X,
                                                const float* __restrict__ W,
                                                float* __restrict__ out, int nTiles) {
    __shared__ float sW[156 * 16];
    for (int i = threadIdx.x; i < 156 * 16; i += 256) sW[i] = W[i];
    __syncthreads();

    const int lane = threadIdx.x & 31;
    const int wave = threadIdx.x >> 5;
    const int n = lane & 15;            // mixture column / sample row in tile
    const bool hi = lane >= 16;
    const int off = hi ? 2 : 0;

    // B layout assumption (mirrors documented A layout): VGPR0 holds K-rows
    // {0|2}, VGPR1 holds {1|3} per lane half; N = lane%16.
    v2f b[39];
#pragma unroll
    for (int s = 0; s < 39; ++s) {
        b[s].x = sW[(4 * s + off) * 16 + n];
        b[s].y = sW[(4 * s + off + 1) * 16 + n];
    }

    const int waveId = blockIdx.x * 8 + wave;
    const int nWaves = gridDim.x * 8;
    for (int tile = waveId; tile < nTiles; tile += nWaves) {
        const float4* row =
            reinterpret_cast<const float4*>(X + (size_t)(tile * 16 + n) * 16);
        float4 r0 = row[0], r1 = row[1], r2 = row[2], r3 = row[3];
        float xv[17];
        xv[0] = r0.x;  xv[1] = r0.y;  xv[2] = r0.z;  xv[3] = r0.w;
        xv[4] = r1.x;  xv[5] = r1.y;  xv[6] = r1.z;  xv[7] = r1.w;
        xv[8] = r2.x;  xv[9] = r2.y;  xv[10] = r2.z; xv[11] = r2.w;
        xv[12] = r3.x; xv[13] = r3.y; xv[14] = r3.z; xv[15] = r3.w;
        xv[16] = 1.f;

        v8f c = {0.f, 0.f, 0.f, 0.f, 0.f, 0.f, 0.f, 0.f};
        c = run_steps(std::make_integer_sequence<int, 39>{}, xv, hi, b, c);

        // C layout: VGPR r -> sample M = r + 8*hi, lane%16 -> mixture.
#pragma unroll
        for (int r = 0; r < 8; ++r) {
            float p = __expf(c[r]);
            p += __shfl_xor(p, 1, 32);
            p += __shfl_xor(p, 2, 32);
            p += __shfl_xor(p, 4, 32);
            p += __shfl_xor(p, 8, 32);
            float res = -__logf(p);
            if (n == r) out[tile * 16 + r + (hi ? 8 : 0)] = res;
        }
    }
}

extern "C" void kernel_launch(void* const* d_in, const int* in_sizes, int n_in,
                              void* d_out, int out_size, void* d_ws, size_t ws_size,
                              hipStream_t stream) {
    const float* samples = (const float*)d_in[0];   // [N,16]
    const float* Phi     = (const float*)d_in[1];   // [16]
    const float* mu      = (const float*)d_in[2];   // [16,16]
    const float* Sigma   = (const float*)d_in[3];   // [16,16,16]
    float* W   = (float*)d_ws;                      // 156*16 floats
    float* out = (float*)d_out;

    int N = in_sizes[0] / 16;
    int nTiles = N / 16;                            // N is a multiple of 16

    gmm_prep<<<1, 32, 0, stream>>>(Phi, mu, Sigma, W);
    int blocks = 512;                               // 4096 waves, ~8 tiles/wave
    gmm_main<<<blocks, 256, 0, stream>>>(samples, W, out, nTiles);
}